// Model_31413390803095
// MI455X (gfx1250) — compile-verified
//
#include <hip/hip_runtime.h>
#include <math.h>

#define Bg    64
#define Nn    1024
#define Hd    128
#define Ecnt  1048576
#define Ccls  6

typedef __attribute__((ext_vector_type(2))) float v2f;
typedef __attribute__((ext_vector_type(8))) float v8f;

// ---------------- utility fills ----------------
__global__ void k_fillf(float* p, float v, int n) {
    int i = blockIdx.x * 256 + threadIdx.x;
    if (i < n) p[i] = v;
}
__global__ void k_filli(int* p, int v, int n) {
    int i = blockIdx.x * 256 + threadIdx.x;
    if (i < n) p[i] = v;
}

// ---------------- edge init (int64 -> int32, mask=1) ----------------
__global__ void k_edges_init(const long long* __restrict__ ei,
                             int* __restrict__ esrc, int* __restrict__ edst,
                             float* __restrict__ emask) {
    int e = blockIdx.x * 256 + threadIdx.x;
    if (e < Ecnt) {
        esrc[e]  = (int)ei[e];
        edst[e]  = (int)ei[Ecnt + e];
        emask[e] = 1.0f;
    }
}

// ---------------- WMMA fp32 GEMM: C[M,128] = A[M,128] @ W[128,128] ----------------
// block = 256 threads = 8 waves; wave w owns N-tile [16w,16w+16); block owns 16 rows.
// Uses V_WMMA_F32_16X16X4_F32 chained over K (32 per tile).
__global__ __launch_bounds__(256) void k_gemm128(const float* __restrict__ A,
                                                 const float* __restrict__ W,
                                                 float* __restrict__ C) {
    __shared__ float xs[16 * 128];   // 8 KB : 16 rows x full K
    __shared__ float wsh[64 * 128];  // 32 KB: K-chunk 64 x N=128
    const int tid  = threadIdx.x;
    const int lane = tid & 31;
    const int wave = tid >> 5;
    const int row0 = blockIdx.x * 16;

    // cooperative load of X tile (512 float4)
    const float4* A4  = (const float4*)(A + (size_t)row0 * 128);
    float4*       xs4 = (float4*)xs;
    for (int i = tid; i < 512; i += 256) xs4[i] = A4[i];

    v8f acc = {};
    const int nb = wave * 16;     // N-tile base
    const int m  = lane & 15;     // A row within tile / C col
    const int hh = lane >> 4;     // lane half selects K pair / M half

    for (int kc = 0; kc < 128; kc += 64) {
        __syncthreads();
        const float4* W4  = (const float4*)(W + (size_t)kc * 128);
        float4*       ws4 = (float4*)wsh;
        for (int i = tid; i < 2048; i += 256) ws4[i] = W4[i];
        __syncthreads();
#pragma unroll
        for (int k0 = 0; k0 < 64; k0 += 4) {
            // A 16x4 f32 fragment: lane half h holds K = k0+2h, k0+2h+1 for row m
            v2f a;
            a.x = xs[m * 128 + kc + k0 + 2 * hh];
            a.y = xs[m * 128 + kc + k0 + 2 * hh + 1];
            // B 4x16 f32 fragment: lane half h holds rows K = k0+2h, k0+2h+1 at col nb+m
            v2f b;
            b.x = wsh[(k0 + 2 * hh) * 128 + nb + m];
            b.y = wsh[(k0 + 2 * hh + 1) * 128 + nb + m];
            acc = __builtin_amdgcn_wmma_f32_16x16x4_f32(
                false, a, false, b, (short)0, acc, false, false);
        }
    }
    // C/D layout: VGPR v, lane half h -> row v + 8h, col = lane&15
    float* cp = C + (size_t)row0 * 128;
#pragma unroll
    for (int v = 0; v < 8; v++)
        cp[(size_t)(v + 8 * hh) * 128 + nb + m] = acc[v];
}

// ---------------- degree accumulation ----------------
__global__ void k_edge_deg(const int* __restrict__ edst, const float* __restrict__ emask,
                           float* __restrict__ deg) {
    int e = blockIdx.x * 256 + threadIdx.x;
    if (e < Ecnt) {
        float m = emask[e];
        if (m > 0.0f) atomicAdd(&deg[edst[e]], m);
    }
}

// ---------------- edge scatter: agg[dst] += h[src] * norm ----------------
// 32 lanes per edge, float4 per lane (128 features)
__global__ void k_edge_scatter(const int* __restrict__ esrc, const int* __restrict__ edst,
                               const float* __restrict__ emask, const float* __restrict__ deg,
                               const float* __restrict__ h, float* __restrict__ agg) {
    unsigned gid = blockIdx.x * 256u + threadIdx.x;
    unsigned e   = gid >> 5;
    unsigned sub = gid & 31;
    float m = emask[e];
    if (m <= 0.0f) return;
    int s = esrc[e], d = edst[e];
    float nrm = m * rsqrtf(deg[s] * deg[d]);
    const float4 v = *(const float4*)(h + (size_t)s * Hd + sub * 4);
    float* ap = agg + (size_t)d * Hd + sub * 4;
    atomicAdd(ap + 0, v.x * nrm);
    atomicAdd(ap + 1, v.y * nrm);
    atomicAdd(ap + 2, v.z * nrm);
    atomicAdd(ap + 3, v.w * nrm);
}

// ---------------- combine: out = agg + h/deg + b ; score = out . ws ; write relu(out) ----------------
__global__ void k_combine(const float* __restrict__ agg_in, const float* __restrict__ h,
                          const float* __restrict__ deg, const float* __restrict__ bias,
                          const float* __restrict__ wsv, float* __restrict__ out,
                          float* __restrict__ score) {
    int i = blockIdx.x;
    int f = threadIdx.x;
    __shared__ float red[Hd];
    float o = agg_in[(size_t)i * Hd + f] + h[(size_t)i * Hd + f] / deg[i] + bias[f];
    red[f] = o * wsv[f];
    out[(size_t)i * Hd + f] = o > 0.0f ? o : 0.0f;
    for (int s = 64; s > 0; s >>= 1) {
        __syncthreads();
        if (f < s) red[f] += red[f + s];
    }
    if (f == 0) score[i] = red[0];
}

// ---------------- per-graph top-k via bitonic sort (descending) ----------------
__global__ __launch_bounds__(1024) void k_topk(const float* __restrict__ score, int n, int k,
                                               int* __restrict__ gidx, float* __restrict__ vals,
                                               int* __restrict__ newpos) {
    __shared__ float ss[1024];
    __shared__ int   si[1024];
    int t = threadIdx.x, g = blockIdx.x;
    ss[t] = score[g * n + t];
    si[t] = t;
    for (int kk = 2; kk <= n; kk <<= 1)
        for (int j = kk >> 1; j > 0; j >>= 1) {
            __syncthreads();
            int ixj = t ^ j;
            if (ixj > t) {
                bool desc = ((t & kk) == 0);
                float a = ss[t], b = ss[ixj];
                bool sw = desc ? (a < b) : (a > b);
                if (sw) {
                    ss[t] = b; ss[ixj] = a;
                    int tmp = si[t]; si[t] = si[ixj]; si[ixj] = tmp;
                }
            }
        }
    __syncthreads();
    if (t < k) {
        int old = g * n + si[t];
        int nid = g * k + t;
        gidx[nid]   = old;
        vals[nid]   = ss[t];
        newpos[old] = nid;
    }
}

// ---------------- gather pooled nodes: x_new = h[gidx] * tanh(val) ----------------
__global__ void k_gather(const float* __restrict__ hin, const int* __restrict__ gidx,
                         const float* __restrict__ vals, float* __restrict__ xout) {
    int nid = blockIdx.x, f = threadIdx.x;
    xout[(size_t)nid * Hd + f] = hin[(size_t)gidx[nid] * Hd + f] * tanhf(vals[nid]);
}

// ---------------- remap edges through newpos, update mask ----------------
__global__ void k_remap(int* __restrict__ esrc, int* __restrict__ edst,
                        float* __restrict__ emask, const int* __restrict__ newpos) {
    int e = blockIdx.x * 256 + threadIdx.x;
    if (e >= Ecnt) return;
    int s = esrc[e], d = edst[e];
    float m = emask[e];
    int ns = newpos[s], nd = newpos[d];
    bool ok = (m > 0.0f) && (ns >= 0) && (nd >= 0);
    esrc[e]  = ns >= 0 ? ns : 0;
    edst[e]  = nd >= 0 ? nd : 0;
    emask[e] = ok ? 1.0f : 0.0f;
}

// ---------------- readout: [max ; mean] per graph -> [B, 256] ----------------
__global__ void k_readout(const float* __restrict__ h, int npg, float inv_n,
                          float* __restrict__ xo) {
    int g = blockIdx.x, f = threadIdx.x;
    float mx = -3.402823466e38f, sm = 0.0f;
    const float* base = h + ((size_t)g * npg) * Hd + f;
    for (int j = 0; j < npg; j++) {
        float v = base[(size_t)j * Hd];
        mx = fmaxf(mx, v);
        sm += v;
    }
    xo[g * 2 * Hd + f]      = mx;
    xo[g * 2 * Hd + Hd + f] = sm * inv_n;
}

// ---------------- z = relu(x1)+relu(x2)+relu(x3) ----------------
__global__ void k_zsum(const float* x1, const float* x2, const float* x3, float* z) {
    int i = blockIdx.x * 256 + threadIdx.x;
    if (i < Bg * 2 * Hd) {
        float a = x1[i], b = x2[i], c = x3[i];
        z[i] = (a > 0.f ? a : 0.f) + (b > 0.f ? b : 0.f) + (c > 0.f ? c : 0.f);
    }
}

// ---------------- MLP head ----------------
__global__ void k_head1(const float* __restrict__ z, const float* __restrict__ W,
                        const float* __restrict__ b, float* __restrict__ o) {
    int g = blockIdx.x, f = threadIdx.x;        // 128 threads
    __shared__ float zr[256];
    zr[f]       = z[g * 256 + f];
    zr[f + 128] = z[g * 256 + 128 + f];
    __syncthreads();
    float acc = b[f];
    for (int k = 0; k < 256; k++) acc += zr[k] * W[k * 128 + f];
    o[g * 128 + f] = acc > 0.f ? acc : 0.f;
}
__global__ void k_head2(const float* __restrict__ z, const float* __restrict__ W,
                        const float* __restrict__ b, float* __restrict__ o) {
    int g = blockIdx.x, f = threadIdx.x;        // 64 threads
    __shared__ float zr[128];
    zr[f]      = z[g * 128 + f];
    zr[f + 64] = z[g * 128 + 64 + f];
    __syncthreads();
    float acc = b[f];
    for (int k = 0; k < 128; k++) acc += zr[k] * W[k * 64 + f];
    o[g * 64 + f] = acc > 0.f ? acc : 0.f;
}
__global__ void k_head3(const float* __restrict__ z, const float* __restrict__ W,
                        const float* __restrict__ b, float* __restrict__ out) {
    int g = blockIdx.x, t = threadIdx.x;        // 64 threads
    __shared__ float zr[64];
    __shared__ float lg[Ccls];
    zr[t] = z[g * 64 + t];
    __syncthreads();
    if (t < Ccls) {
        float acc = b[t];
        for (int k = 0; k < 64; k++) acc += zr[k] * W[k * Ccls + t];
        lg[t] = acc;
    }
    __syncthreads();
    if (t == 0) {
        float m = lg[0];
        for (int c = 1; c < Ccls; c++) m = fmaxf(m, lg[c]);
        float s = 0.f;
        for (int c = 0; c < Ccls; c++) s += expf(lg[c] - m);
        float lse = m + logf(s);
        for (int c = 0; c < Ccls; c++) out[g * Ccls + c] = lg[c] - lse;
    }
}

// ---------------- launch ----------------
extern "C" void kernel_launch(void* const* d_in, const int* in_sizes, int n_in,
                              void* d_out, int out_size, void* d_ws, size_t ws_size,
                              hipStream_t stream) {
    const float*     x   = (const float*)d_in[0];
    const long long* ei  = (const long long*)d_in[1];
    const float* W1 = (const float*)d_in[3],  *b1 = (const float*)d_in[4],  *s1 = (const float*)d_in[5];
    const float* W2 = (const float*)d_in[6],  *b2 = (const float*)d_in[7],  *s2 = (const float*)d_in[8];
    const float* W3 = (const float*)d_in[9],  *b3 = (const float*)d_in[10], *s3 = (const float*)d_in[11];
    const float* L1w = (const float*)d_in[12], *L1b = (const float*)d_in[13];
    const float* L2w = (const float*)d_in[14], *L2b = (const float*)d_in[15];
    const float* L3w = (const float*)d_in[16], *L3b = (const float*)d_in[17];
    float* out = (float*)d_out;

    char* base = (char*)d_ws;
    size_t off = 0;
    auto carve = [&](size_t bytes) -> void* {
        void* q = base + off;
        off += (bytes + 255) & ~(size_t)255;
        return q;
    };
    int*   esrc   = (int*)carve((size_t)Ecnt * 4);
    int*   edst   = (int*)carve((size_t)Ecnt * 4);
    float* emask  = (float*)carve((size_t)Ecnt * 4);
    float* deg    = (float*)carve((size_t)65536 * 4);
    float* score  = (float*)carve((size_t)65536 * 4);
    float* vals   = (float*)carve((size_t)32768 * 4);
    int*   gidx   = (int*)carve((size_t)32768 * 4);
    int*   newpos = (int*)carve((size_t)65536 * 4);
    float* hb     = (float*)carve((size_t)65536 * 128 * 4);
    float* ag     = (float*)carve((size_t)65536 * 128 * 4);
    float* xp     = (float*)carve((size_t)32768 * 128 * 4);
    float* x1     = (float*)carve((size_t)Bg * 256 * 4);
    float* x2     = (float*)carve((size_t)Bg * 256 * 4);
    float* x3     = (float*)carve((size_t)Bg * 256 * 4);
    float* zb     = (float*)carve((size_t)Bg * 256 * 4);
    float* z1     = (float*)carve((size_t)Bg * 128 * 4);
    float* z2     = (float*)carve((size_t)Bg * 64 * 4);

    k_edges_init<<<Ecnt / 256, 256, 0, stream>>>(ei, esrc, edst, emask);

    // ---- layer 1: 65536 nodes -> pool to 32768 ----
    k_gemm128<<<65536 / 16, 256, 0, stream>>>(x, W1, hb);
    k_fillf<<<(65536 + 255) / 256, 256, 0, stream>>>(deg, 1.0f, 65536);
    k_edge_deg<<<Ecnt / 256, 256, 0, stream>>>(edst, emask, deg);
    k_fillf<<<(65536 * 128 + 255) / 256, 256, 0, stream>>>(ag, 0.0f, 65536 * 128);
    k_edge_scatter<<<Ecnt / 8, 256, 0, stream>>>(esrc, edst, emask, deg, hb, ag);
    k_combine<<<65536, 128, 0, stream>>>(ag, hb, deg, b1, s1, ag, score);
    k_filli<<<(65536 + 255) / 256, 256, 0, stream>>>(newpos, -1, 65536);
    k_topk<<<Bg, 1024, 0, stream>>>(score, 1024, 512, gidx, vals, newpos);
    k_gather<<<32768, 128, 0, stream>>>(ag, gidx, vals, xp);
    k_remap<<<Ecnt / 256, 256, 0, stream>>>(esrc, edst, emask, newpos);
    k_readout<<<Bg, 128, 0, stream>>>(xp, 512, 1.0f / 512.0f, x1);

    // ---- layer 2: 32768 nodes -> pool to 16384 ----
    k_gemm128<<<32768 / 16, 256, 0, stream>>>(xp, W2, hb);
    k_fillf<<<(32768 + 255) / 256, 256, 0, stream>>>(deg, 1.0f, 32768);
    k_edge_deg<<<Ecnt / 256, 256, 0, stream>>>(edst, emask, deg);
    k_fillf<<<(32768 * 128 + 255) / 256, 256, 0, stream>>>(ag, 0.0f, 32768 * 128);
    k_edge_scatter<<<Ecnt / 8, 256, 0, stream>>>(esrc, edst, emask, deg, hb, ag);
    k_combine<<<32768, 128, 0, stream>>>(ag, hb, deg, b2, s2, ag, score);
    k_filli<<<(32768 + 255) / 256, 256, 0, stream>>>(newpos, -1, 32768);
    k_topk<<<Bg, 512, 0, stream>>>(score, 512, 256, gidx, vals, newpos);
    k_gather<<<16384, 128, 0, stream>>>(ag, gidx, vals, xp);
    k_remap<<<Ecnt / 256, 256, 0, stream>>>(esrc, edst, emask, newpos);
    k_readout<<<Bg, 128, 0, stream>>>(xp, 256, 1.0f / 256.0f, x2);

    // ---- layer 3: 16384 nodes, no pool ----
    k_gemm128<<<16384 / 16, 256, 0, stream>>>(xp, W3, hb);
    k_fillf<<<(16384 + 255) / 256, 256, 0, stream>>>(deg, 1.0f, 16384);
    k_edge_deg<<<Ecnt / 256, 256, 0, stream>>>(edst, emask, deg);
    k_fillf<<<(16384 * 128 + 255) / 256, 256, 0, stream>>>(ag, 0.0f, 16384 * 128);
    k_edge_scatter<<<Ecnt / 8, 256, 0, stream>>>(esrc, edst, emask, deg, hb, ag);
    k_combine<<<16384, 128, 0, stream>>>(ag, hb, deg, b3, s3, ag, score);
    k_readout<<<Bg, 128, 0, stream>>>(ag, 256, 1.0f / 256.0f, x3);

    // ---- head ----
    k_zsum<<<(Bg * 256 + 255) / 256, 256, 0, stream>>>(x1, x2, x3, zb);
    k_head1<<<Bg, 128, 0, stream>>>(zb, L1w, L1b, z1);
    k_head2<<<Bg, 64, 0, stream>>>(z1, L2w, L2b, z2);
    k_head3<<<Bg, 64, 0, stream>>>(z2, L3w, L3b, out);
}